// OptimizedUniformSampler_40321152974972
// MI455X (gfx1250) — compile-verified
//
#include <hip/hip_runtime.h>
#include <hip/hip_bf16.h>
#include <stdint.h>

// Negative sampler + sorted-hash triple filter for gfx1250 (MI455X).
//
// Strategy (latency-bound binary search, no matrix math => no WMMA):
//  - kernel 1: build a 2^K-1 entry "uniform binary search" pivot tree
//    (level-order) in d_ws, with INT64_MAX sentinels for out-of-range nodes.
//  - kernel 2: each 256-thread block async-copies the 32KB pivot tree into
//    LDS (CDNA5 GLOBAL_LOAD_ASYNC_TO_LDS_B64 + s_wait_asynccnt), processes
//    2048 rows: corrupt triple, hash, 12 LDS levels + ~8 global levels of
//    lower_bound, write 3 floats + keep flag.

typedef long long i64;
typedef int v2i_t __attribute__((vector_size(2 * sizeof(int))));
typedef __attribute__((address_space(1))) v2i_t* gv2i_p;   // global
typedef __attribute__((address_space(3))) v2i_t* lv2i_p;   // LDS

#define THREADS 256
#define ITEMS   8      // rows per thread -> 2048 rows per block
#define MAXK    12     // 2^12 * 8B = 32KB LDS pivot cache

// ---------------------------------------------------------------------------
// Kernel 1: build pivot tree. Node t (1-based) at level j = floor(log2 t)
// with path bits b_0..b_{j-1} (MSB-first) corresponds to search state
// pos = sum b_i*(S>>i); it probes hashes[pos + (S>>j) - 1] guarded by
// pos + (S>>j) <= L. Out-of-range nodes store INT64_MAX (hashes < 2^59, so
// the sentinel never advances pos -> identical to the guarded search).
// ---------------------------------------------------------------------------
__global__ void build_pivot_tree(const i64* __restrict__ hashes,
                                 i64* __restrict__ tab,
                                 unsigned L, unsigned S, int K) {
  unsigned n = 1u << K;
  for (unsigned t = blockIdx.x * blockDim.x + threadIdx.x; t < n;
       t += gridDim.x * blockDim.x) {
    i64 v;
    if (t == 0u) {
      v = 0;  // unused slot
    } else {
      int j = 31 - __builtin_clz(t);
      unsigned path = t - (1u << j);
      unsigned pos = 0;
      for (int i = 0; i < j; ++i)
        if ((path >> (j - 1 - i)) & 1u) pos += (S >> i);
      unsigned nxt = pos + (S >> j);
      v = (nxt <= L) ? hashes[nxt - 1u] : (i64)0x7FFFFFFFFFFFFFFFLL;
    }
    tab[t] = v;
  }
}

// ---------------------------------------------------------------------------
// Kernel 2: sample + filter.
// ---------------------------------------------------------------------------
__global__ __launch_bounds__(THREADS) void neg_sample_filter(
    const i64* __restrict__ posb,    // [B,3]
    const i64* __restrict__ randv,   // [total]
    const i64* __restrict__ hashes,  // [L] sorted unique
    const i64* __restrict__ pivots,  // [1<<K] level-order pivot tree
    float* __restrict__ out,         // [3*total] neg, then [total] keep
    unsigned total, unsigned split, unsigned numNegs, int negShift,
    unsigned L, unsigned S, int K) {
  extern __shared__ i64 smem[];
  const unsigned tid = threadIdx.x;

  if (K > 0) {
    const unsigned nT = 1u << K;
#if defined(__HIP_DEVICE_COMPILE__) && \
    __has_builtin(__builtin_amdgcn_global_load_async_to_lds_b64)
    // CDNA5 async copy: per-lane 8B global->LDS, tracked by ASYNCcnt.
    for (unsigned i = tid; i < nT; i += THREADS) {
      __builtin_amdgcn_global_load_async_to_lds_b64(
          (gv2i_p)(pivots + i), (lv2i_p)(smem + i), 0, 0);
    }
#if __has_builtin(__builtin_amdgcn_s_wait_asynccnt)
    __builtin_amdgcn_s_wait_asynccnt(0);
#else
    asm volatile("s_wait_asynccnt 0" ::: "memory");
#endif
#else
    for (unsigned i = tid; i < nT; i += THREADS) smem[i] = pivots[i];
#endif
    __syncthreads();
  }

  const unsigned base = blockIdx.x * (THREADS * ITEMS) + tid;
  for (int it = 0; it < ITEMS; ++it) {
    unsigned row = base + (unsigned)it * THREADS;
    if (row >= total) return;

    // Prefetch next sweep's rand value (speculative: OOB silently dropped).
    __builtin_prefetch(randv + row + THREADS, 0, 1);

    // One positive triple per wave32 (64 negs/pos => wave-uniform broadcast).
    unsigned b = (negShift >= 0) ? (row >> negShift) : (row / numNegs);
    i64 e0 = posb[3u * b + 0];
    i64 rr = posb[3u * b + 1];
    i64 e2 = posb[3u * b + 2];
    i64 rv = randv[row];

    bool ch = row < split;                 // corrupt head else tail
    i64 orig = ch ? e0 : e2;
    i64 repl = rv + (((rv >= orig) && (orig > 0)) ? (i64)1 : (i64)0);
    i64 hN = ch ? repl : e0;
    i64 tN = ch ? e2 : repl;
    i64 h = (hN << 42) | (rr << 21) | tN;  // < 2^59, positive

    // --- lower_bound: K levels from LDS, remainder from global (L2-hot) ---
    unsigned pos = 0;
    unsigned node = 1;
    for (int j = 0; j < K; ++j) {
      bool adv = smem[node] < h;           // sentinel handles nxt > L
      if (adv) pos += (S >> j);
      node = (node << 1) | (adv ? 1u : 0u);
    }
    for (unsigned step = (S >> K); step; step >>= 1) {
      unsigned nxt = pos + step;
      unsigned idx = (nxt <= L) ? (nxt - 1u) : (L - 1u);  // clamped load
      i64 v = hashes[idx];
      if ((nxt <= L) && (v < h)) pos = nxt;
    }
    bool inset = (pos < L) && (hashes[(pos < L) ? pos : 0u] == h);

    // --- outputs: coalesced 12B/lane triple + keep flag ---
    size_t o = (size_t)3 * row;
    out[o + 0] = (float)hN;   // values <= 100000 < 2^24: exact in f32
    out[o + 1] = (float)rr;
    out[o + 2] = (float)tN;
    out[(size_t)3 * total + row] = inset ? 0.0f : 1.0f;
  }
}

// ---------------------------------------------------------------------------
extern "C" void kernel_launch(void* const* d_in, const int* in_sizes, int n_in,
                              void* d_out, int out_size, void* d_ws,
                              size_t ws_size, hipStream_t stream) {
  const i64* posb   = (const i64*)d_in[0];
  const i64* randv  = (const i64*)d_in[1];
  const i64* hashes = (const i64*)d_in[2];
  // d_in[3] (num_negs scalar) derived from sizes instead (no device read).

  unsigned B = (unsigned)(in_sizes[0] / 3);
  unsigned total = (unsigned)in_sizes[1];
  unsigned L = (unsigned)in_sizes[2];
  unsigned numNegs = B ? (total / B) : 64u;
  unsigned split = (total + 1u) / 2u;   // ceil(total/2)

  int tlev = 31 - __builtin_clz(L | 1u);    // floor(log2 L)
  unsigned S = 1u << tlev;                  // initial search step
  int K = (tlev < MAXK) ? tlev : MAXK;      // LDS-cached levels
  while (K > 0 && (((size_t)8u) << K) > ws_size) --K;  // fit in workspace

  i64* pivots = (i64*)d_ws;
  if (K > 0) {
    unsigned n = 1u << K;
    build_pivot_tree<<<(n + THREADS - 1) / THREADS, THREADS, 0, stream>>>(
        hashes, pivots, L, S, K);
  }

  int negShift = (numNegs && ((numNegs & (numNegs - 1u)) == 0u))
                     ? __builtin_ctz(numNegs)
                     : -1;

  unsigned rowsPerBlk = THREADS * ITEMS;
  unsigned blocks = (total + rowsPerBlk - 1u) / rowsPerBlk;
  size_t smemBytes = ((size_t)8u) << (K > 0 ? K : 0);

  neg_sample_filter<<<blocks, THREADS, smemBytes, stream>>>(
      posb, randv, hashes, pivots, (float*)d_out, total, split, numNegs,
      negShift, L, S, K);
}